// losses_22668837388898
// MI455X (gfx1250) — compile-verified
//
#include <hip/hip_runtime.h>

#define EPS7 1e-7f
#define TILE 1024
#define BLK  256

#define FSQRT(x) __builtin_amdgcn_sqrtf(x)
#define FRCP(x)  __builtin_amdgcn_rcpf(x)

__device__ __forceinline__ void sigma_of(float w, float h, float r,
                                         float& s11, float& s12, float& s22,
                                         float& det) {
  w = fminf(fmaxf(w, 1e-7f), 1e7f);
  h = fminf(fmaxf(h, 1e-7f), 1e7f);
  float ha = 0.5f * w, hb = 0.5f * h;
  float a = ha * ha, b = hb * hb;
  float s, c;
  __sincosf(r, &s, &c);
  float cc = c * c, ss = s * s, sc = s * c;
  s11 = a * cc + b * ss;
  s12 = (a - b) * sc;
  s22 = a * ss + b * cc;
  det = a * b;
}

__device__ __forceinline__ float postproc(float d) {
  // log1p then 1 - 1/(1+l)  ==  l/(1+l)
  float l = log1pf(d);
  return l * FRCP(1.0f + l);
}

__device__ __forceinline__ void compute3(float pw, float ph, float pr,
                                         float tw, float th, float trr,
                                         float& gwd, float& kfi, float& kld) {
  float p11, p12, p22, dp, t11, t12, t22, dt;
  sigma_of(pw, ph, pr, p11, p12, p22, dp);
  sigma_of(tw, th, trr, t11, t12, t22, dt);

  // ---------------- GWD ----------------
  {
    float tr    = (p11 + p22) + (t11 + t22);
    float t_tr  = p11 * t11 + 2.0f * p12 * t12 + p22 * t22;
    float det_s = FSQRT(fmaxf(dp * dt, EPS7));
    float whr   = tr - 2.0f * FSQRT(fmaxf(t_tr + 2.0f * det_s, EPS7));
    float dist  = FSQRT(fmaxf(whr, EPS7));
    float scale = 2.0f * fmaxf(FSQRT(fmaxf(FSQRT(fmaxf(det_s, EPS7)), EPS7)), EPS7);
    gwd = postproc(dist * FRCP(scale));
  }
  // ---------------- KFIoU ----------------
  {
    float vb_p = 4.0f * FSQRT(dp);
    float vb_t = 4.0f * FSQRT(dt);
    float m11 = p11 + t11, m12 = p12 + t12, m22 = p22 + t22;
    float dsum = m11 * m22 - m12 * m12;
    float inv  = FRCP(dsum);
    float k11 = (p11 * m22 - p12 * m12) * inv;
    float k12 = (p12 * m11 - p11 * m12) * inv;
    float k21 = (p12 * m22 - p22 * m12) * inv;
    float k22 = (p22 * m11 - p12 * m12) * inv;
    float g11 = p11 - (k11 * p11 + k12 * p12);
    float g12 = p12 - (k11 * p12 + k12 * p22);
    float g21 = p12 - (k21 * p11 + k22 * p12);
    float g22 = p22 - (k21 * p12 + k22 * p22);
    float dg = g11 * g22 - g12 * g21;
    float vb = 4.0f * FSQRT(fmaxf(dg, 1e-6f));
    if (vb != vb) vb = 0.0f;
    float kfiou = vb * FRCP(vb_p + vb_t - vb + 1e-6f);
    kfi = fmaxf(1.0f - kfiou, 0.0f);
  }
  // ---------------- KLD ----------------
  {
    float trc  = (p22 * t11 - 2.0f * p12 * t12 + p11 * t22) * FRCP(dp);
    float whr  = 0.5f * trc + 0.5f * (__logf(dp) - __logf(dt)) - 1.0f;
    float dist = FSQRT(fmaxf(whr, EPS7));
    kld = postproc(dist);
  }
}

__global__ __launch_bounds__(BLK) void losses_22668837388898_kernel(
    const float* __restrict__ pred, const float* __restrict__ targ,
    float* __restrict__ out, long long n) {
  __shared__ float sp[TILE * 3];
  __shared__ float st[TILE * 3];
  const int tid = threadIdx.x;
  const long long tileBase = (long long)blockIdx.x * TILE;

  if (tileBase + TILE <= n) {
    // ---- stage tile into LDS with CDNA5 async global->LDS b128 loads ----
    unsigned lds_p = (unsigned)(size_t)(&sp[0]);
    unsigned lds_t = (unsigned)(size_t)(&st[0]);
    const float* gp = pred + tileBase * 3;   // 16B-aligned (tileBase*12 % 16 == 0)
    const float* gt = targ + tileBase * 3;
#pragma unroll
    for (int kk = 0; kk < 3; ++kk) {
      unsigned off = (unsigned)(kk * (BLK * 16) + tid * 16);  // covers TILE*12 bytes
      unsigned lp = lds_p + off;
      unsigned lt = lds_t + off;
      asm volatile("global_load_async_to_lds_b128 %0, %1, %2"
                   :: "v"(lp), "v"(off), "s"(gp) : "memory");
      asm volatile("global_load_async_to_lds_b128 %0, %1, %2"
                   :: "v"(lt), "v"(off), "s"(gt) : "memory");
    }
    asm volatile("s_wait_asynccnt 0x0" ::: "memory");
    __syncthreads();

#pragma unroll
    for (int it = 0; it < TILE / BLK; ++it) {
      int e = tid + it * BLK;
      float pw = sp[e * 3 + 0], ph = sp[e * 3 + 1], pr = sp[e * 3 + 2];
      float tw = st[e * 3 + 0], th = st[e * 3 + 1], tr = st[e * 3 + 2];
      float gwd, kfi, kld;
      compute3(pw, ph, pr, tw, th, tr, gwd, kfi, kld);
      long long gi = tileBase + e;
      out[gi]         = gwd;
      out[n + gi]     = kfi;
      out[2 * n + gi] = kld;
    }
  } else {
    // tail tile: guarded scalar path
    for (int it = 0; it < TILE / BLK; ++it) {
      long long e = tileBase + tid + it * BLK;
      if (e < n) {
        const float* P = pred + e * 3;
        const float* T = targ + e * 3;
        float gwd, kfi, kld;
        compute3(P[0], P[1], P[2], T[0], T[1], T[2], gwd, kfi, kld);
        out[e]         = gwd;
        out[n + e]     = kfi;
        out[2 * n + e] = kld;
      }
    }
  }
}

extern "C" void kernel_launch(void* const* d_in, const int* in_sizes, int n_in,
                              void* d_out, int out_size, void* d_ws, size_t ws_size,
                              hipStream_t stream) {
  const float* pred = (const float*)d_in[0];
  const float* targ = (const float*)d_in[1];
  float* out = (float*)d_out;
  long long n = (long long)in_sizes[0] / 3;
  int blocks = (int)((n + TILE - 1) / TILE);
  losses_22668837388898_kernel<<<blocks, BLK, 0, stream>>>(pred, targ, out, n);
}